// TimeInvariantFilter_22170621182383
// MI455X (gfx1250) — compile-verified
//
#include <hip/hip_runtime.h>
#include <hip/hip_bf16.h>
#include <math.h>

// ---------------------------------------------------------------------------
// Fused TimeInvariantFilter for gfx1250 (MI455X, wave32, WMMA + TDM).
//
// Pipeline: complex beamform -> |.| -> GEMM(F->K) via V_WMMA_F32_16X16X4_F32
//           -> relu -> log -> (deterministic 2-pass) BatchNorm -> transpose.
// HBM-bound (~1.1 GB traffic, ~5.3 GFLOP) => fuse to avoid materializing mag.
// w_proj (80x257 f32, 82 KB) is DMA'd once per workgroup into LDS with the
// Tensor Data Mover, so the WMMA inner loop reads only LDS.
// ---------------------------------------------------------------------------

typedef float v2f __attribute__((ext_vector_type(2)));
typedef float v8f __attribute__((ext_vector_type(8)));
typedef unsigned int u32x4 __attribute__((ext_vector_type(4)));
typedef int i32x4 __attribute__((ext_vector_type(4)));
typedef int i32x8 __attribute__((ext_vector_type(8)));

namespace tif {
constexpr int N = 16;
constexpr int C = 4;
constexpr int F = 257;
constexpr int T = 1000;
constexpr int B = 8;
constexpr int K = 80;

constexpr int TM      = 16;                    // t-rows per workgroup (WMMA M)
constexpr int TILES_T = (T + TM - 1) / TM;     // 63
constexpr int CF      = 32;                    // f-bins staged in LDS per chunk
constexpr int NCH     = (F + CF - 1) / CF;     // 9
constexpr int NWG     = N * B * TILES_T;       // 8064
constexpr int KT      = K / 16;                // 5 K-tiles (waves 0..4 do WMMA)

constexpr float EPS    = 1e-5f;
constexpr float BN_EPS = 1e-5f;
}

// ---------------------------------------------------------------------------
// Kernel 1: per (n, b, t-tile):
//   TDM: w_proj -> LDS (once);  then per 32-f chunk: beamform+mag -> LDS,
//   WMMA GEMM over F, finally relu+log -> g[n][b][t][k] + BN partial sums.
// ---------------------------------------------------------------------------
__global__ __launch_bounds__(256)
void tif_fused_wmma_kernel(const float* __restrict__ xr,
                           const float* __restrict__ xi,
                           const float* __restrict__ wr,
                           const float* __restrict__ wi,
                           const float* __restrict__ wproj,
                           float* __restrict__ g,
                           float* __restrict__ partial) {
  using namespace tif;
  __shared__ float Wlds[K * F];    // w_proj row-major [k][f], filled by TDM
  __shared__ float Amat[CF][TM];   // mag tile: [f-in-chunk][t-row]
  __shared__ float red1[8], red2[8];

  const int wg  = blockIdx.x;
  const int tt  = wg % TILES_T;
  const int b   = (wg / TILES_T) % B;
  const int n   = wg / (TILES_T * B);
  const int t0  = tt * TM;
  const int tid  = threadIdx.x;
  const int lane = tid & 31;
  const int wave = tid >> 5;

  // ---- TDM: DMA the full w_proj matrix (80 x 257 f32) into LDS -----------
  // D# per CDNA5 ISA ch.8: group0 = {count, lds_addr, global_addr, type=2},
  // group1 = {data_size=4B, dims/strides/tile}, groups 2/3 zero (2-D tensor).
  if (wave == 0) {
    const unsigned long long ga = (unsigned long long)(uintptr_t)wproj;
    // LDS aperture maps flat addr[31:0] -> LDS byte offset (ISA 10.2).
    const unsigned int lds_base = (unsigned int)(uintptr_t)(void*)&Wlds[0];
    u32x4 g0;
    g0[0] = 1u;                                      // count=1 (valid user D#)
    g0[1] = lds_base;                                // lds_addr (bytes)
    g0[2] = (unsigned int)(ga & 0xFFFFFFFFu);        // global_addr[31:0]
    g0[3] = (unsigned int)((ga >> 32) & 0x1FFFFFFu)  // global_addr[56:32]
          | (2u << 30);                              // type = 2 ("image")
    i32x8 g1;
    g1[0] = (int)(2u << 16);                         // data_size = 4 bytes
    g1[1] = (int)((unsigned)F << 16);                // tensor_dim0[15:0]
    g1[2] = (int)((unsigned)K << 16);                // dim0 hi=0 | tensor_dim1 lo=K
    g1[3] = (int)((unsigned)F << 16);                // dim1 hi=0 | tile_dim0=F
    g1[4] = (int)K;                                  // tile_dim1=K | tile_dim2=0
    g1[5] = (int)F;                                  // tensor_dim0_stride = F
    g1[6] = 0;                                       // stride hi | dim1_stride lo
    g1[7] = 0;
    i32x4 gz = {0, 0, 0, 0};
#if __clang_major__ >= 23
    i32x8 gz8 = {};
    __builtin_amdgcn_tensor_load_to_lds(g0, g1, gz, gz, gz8, 0);
#else
    __builtin_amdgcn_tensor_load_to_lds(g0, g1, gz, gz, 0);
#endif
    __builtin_amdgcn_s_wait_tensorcnt(0);
  }
  __syncthreads();

  const int k0  = wave * 16;       // K-tile owned by waves 0..4
  v8f acc = {};                    // 16x16 f32 accumulator

  for (int ch = 0; ch < NCH; ++ch) {
    const int fbase = ch * CF;

    // ---- stage mag tile into LDS: 32f x 16t, 2 elements per thread --------
    for (int e = tid; e < CF * TM; e += 256) {
      const int fi = e >> 4;       // f within chunk
      const int ti = e & 15;       // t within tile (contiguous -> coalesced x)
      const int f  = fbase + fi;
      const int t  = t0 + ti;
      float m = 0.0f;
      if (f < F && t < T) {
        float br = 0.0f, bi = 0.0f;
        const size_t xoff = ((size_t)n * C * F + (size_t)f) * T + (size_t)t;
        const size_t woff = ((size_t)f * B + (size_t)b) * C;
#pragma unroll
        for (int c = 0; c < C; ++c) {
          const float a  = xr[xoff + (size_t)c * F * T];
          const float d  = xi[xoff + (size_t)c * F * T];
          const float pr = wr[woff + c];
          const float pi = wi[woff + c];
          br += a * pr - d * pi;
          bi += d * pr + a * pi;
        }
        m = sqrtf(br * br + bi * bi + EPS);
      }
      Amat[fi][ti] = m;
    }
    __syncthreads();

    // ---- WMMA: waves 0..4, 8 x K=4 steps over this 32-f chunk -------------
    if (wave < KT) {
      const int m16 = lane & 15;   // A: row M / B: column N
      const int kp  = lane >> 4;   // K-pair select (0 -> K{0,1}, 1 -> K{2,3})
      const int kcol = k0 + m16;
#pragma unroll
      for (int kk = 0; kk < 8; ++kk) {
        const int fo = kk * 4;
        v2f afrag, bfrag;
        afrag.x = Amat[fo + 2 * kp + 0][m16];
        afrag.y = Amat[fo + 2 * kp + 1][m16];
        const int f0 = fbase + fo + 2 * kp;
        bfrag.x = (f0 + 0 < F) ? Wlds[kcol * F + f0 + 0] : 0.0f;
        bfrag.y = (f0 + 1 < F) ? Wlds[kcol * F + f0 + 1] : 0.0f;
        acc = __builtin_amdgcn_wmma_f32_16x16x4_f32(
            /*neg_a=*/false, afrag, /*neg_b=*/false, bfrag,
            /*c_mod=*/(short)0, acc, /*reuse_a=*/false, /*reuse_b=*/false);
      }
    }
    __syncthreads();
  }

  // ---- epilogue: relu + log, store g, accumulate BN partials --------------
  float s1 = 0.0f, s2 = 0.0f;
  if (wave < KT) {
    const int col  = lane & 15;    // N index (k)
    const int half = lane >> 4;    // M block select
#pragma unroll
    for (int r = 0; r < 8; ++r) {
      const int row = r + half * 8;        // M index (t within tile)
      const int t   = t0 + row;
      float v = acc[r];
      v = logf(fmaxf(v, 0.0f) + EPS);
      if (t < T) {
        g[(((size_t)n * B + b) * T + t) * K + (k0 + col)] = v;
        s1 += v;
        s2 += v * v;
      }
    }
  }
  // wave32 tree reduce
#pragma unroll
  for (int off = 16; off > 0; off >>= 1) {
    s1 += __shfl_down(s1, off, 32);
    s2 += __shfl_down(s2, off, 32);
  }
  if (lane == 0) { red1[wave] = s1; red2[wave] = s2; }
  __syncthreads();
  if (tid == 0) {
    float a1 = 0.0f, a2 = 0.0f;
#pragma unroll
    for (int w = 0; w < KT; ++w) { a1 += red1[w]; a2 += red2[w]; }
    partial[2 * wg + 0] = a1;
    partial[2 * wg + 1] = a2;
  }
}

// ---------------------------------------------------------------------------
// Kernel 2: deterministic reduction of per-workgroup partials -> scale/shift
// per channel b.  scale = gamma*rsqrt(var+eps), shift = beta - mean*scale.
// ---------------------------------------------------------------------------
__global__ __launch_bounds__(256)
void tif_bn_stats_kernel(const float* __restrict__ partial,
                         const float* __restrict__ gamma,
                         const float* __restrict__ beta,
                         float* __restrict__ sb) {
  using namespace tif;
  const int b   = blockIdx.x;
  const int tid = threadIdx.x;
  float s1 = 0.0f, s2 = 0.0f;
  for (int i = tid; i < N * TILES_T; i += 256) {
    const int nn = i / TILES_T;
    const int tt = i % TILES_T;
    const int wg = (nn * B + b) * TILES_T + tt;
    s1 += partial[2 * wg + 0];
    s2 += partial[2 * wg + 1];
  }
  __shared__ float r1[256], r2[256];
  r1[tid] = s1;
  r2[tid] = s2;
  __syncthreads();
  for (int o = 128; o > 0; o >>= 1) {
    if (tid < o) { r1[tid] += r1[tid + o]; r2[tid] += r2[tid + o]; }
    __syncthreads();
  }
  if (tid == 0) {
    const float cnt  = (float)N * (float)T * (float)K;
    const float mean = r1[0] / cnt;
    const float var  = r2[0] / cnt - mean * mean;
    const float scale = gamma[b] * rsqrtf(var + BN_EPS);
    sb[2 * b + 0] = scale;
    sb[2 * b + 1] = beta[b] - mean * scale;
  }
}

// ---------------------------------------------------------------------------
// Kernel 3: normalize + transpose (N,B,T,K) -> (N,T,B*K).  Fully coalesced:
// consecutive output elements are consecutive k, which is also contiguous in g.
// ---------------------------------------------------------------------------
__global__ __launch_bounds__(256)
void tif_bn_apply_kernel(const float* __restrict__ g,
                         const float* __restrict__ sb,
                         float* __restrict__ out) {
  using namespace tif;
  const size_t total = (size_t)N * T * B * K;
  const size_t i = (size_t)blockIdx.x * 256 + threadIdx.x;
  if (i >= total) return;
  const int k = (int)(i % K);
  const int b = (int)((i / K) % B);
  const int t = (int)((i / ((size_t)K * B)) % T);
  const int n = (int)(i / ((size_t)K * B * T));
  const float v = g[(((size_t)n * B + b) * T + t) * K + k];
  out[i] = v * sb[2 * b + 0] + sb[2 * b + 1];
}

// ---------------------------------------------------------------------------
extern "C" void kernel_launch(void* const* d_in, const int* in_sizes, int n_in,
                              void* d_out, int out_size, void* d_ws, size_t ws_size,
                              hipStream_t stream) {
  using namespace tif;
  const float* xr    = (const float*)d_in[0];
  const float* xi    = (const float*)d_in[1];
  const float* wr    = (const float*)d_in[2];
  const float* wi    = (const float*)d_in[3];
  const float* wproj = (const float*)d_in[4];
  const float* gamma = (const float*)d_in[5];
  const float* beta  = (const float*)d_in[6];
  float* out = (float*)d_out;

  // workspace layout
  float* g       = (float*)d_ws;                        // N*B*T*K floats (~41 MB)
  float* partial = g + (size_t)N * B * T * K;           // 2*NWG floats
  float* sb      = partial + (size_t)2 * NWG;           // 2*B floats

  tif_fused_wmma_kernel<<<NWG, 256, 0, stream>>>(xr, xi, wr, wi, wproj, g, partial);
  tif_bn_stats_kernel<<<B, 256, 0, stream>>>(partial, gamma, beta, sb);

  const size_t total  = (size_t)N * T * B * K;
  const int    blocks = (int)((total + 255) / 256);
  tif_bn_apply_kernel<<<blocks, 256, 0, stream>>>(g, sb, out);
}